// SimpleAGG_78907139162590
// MI455X (gfx1250) — compile-verified
//
#include <hip/hip_runtime.h>

// ----------------------------------------------------------------------------
// 3-hop GraphSAGE (F=1, aggregator='add', bias=False) for MI455X / gfx1250.
//   per hop:  hout[i]   = w_self[hop]  * hin[i]                (init kernel)
//             hout[dst] += w_neigh[hop] * hin[src]  per edge   (scatter kernel)
// Memory-bound scatter-add: 256MB/hop streaming indices via async->LDS (NT),
// h gather + fp32 atomics both L2-resident (4MB each vs 192MB L2).
// Hardware fp32 no-return atomics forced via inline asm (no CAS loop).
// ----------------------------------------------------------------------------

#define WAVES_PER_BLOCK 8
#define THREADS (WAVES_PER_BLOCK * 32)
#define EDGES_PER_STAGE 128   // 128 edges * 4B = 512B = 32 lanes * 16B (one b128/lane)
#define STAGE_BYTES 512

// LDS staging: [2 buffers][wave][src|dst][lane] of int4  -> 16 KB / block
struct StageBuf { int4 d[2][WAVES_PER_BLOCK][2][32]; };

__device__ __forceinline__ void async_ld_b128(unsigned lds_off, const void* gaddr) {
  asm volatile("global_load_async_to_lds_b128 %0, %1, off th:TH_LOAD_NT"
               :: "v"(lds_off), "v"(gaddr) : "memory");
}

__device__ __forceinline__ void atomic_add_f32_dev(float* p, float v) {
  // No-return global fp32 atomic add, device scope (visible across WGPs).
  // S_ENDPGM implicitly waits all counters, so stores drain before kernel end.
  asm volatile("global_atomic_add_f32 %0, %1, off scope:SCOPE_DEV"
               :: "v"(p), "v"(v) : "memory");
}

__global__ void __launch_bounds__(THREADS)
sage_init_scale(const float* __restrict__ hin, float* __restrict__ hout,
                const float* __restrict__ w_self, int hop, int n) {
  const float ws = w_self[hop];           // uniform -> s_load
  int i = (blockIdx.x * blockDim.x + threadIdx.x) * 4;
  if (i + 3 < n) {
    float4 v = *(const float4*)(hin + i);
    v.x *= ws; v.y *= ws; v.z *= ws; v.w *= ws;
    *(float4*)(hout + i) = v;
  } else {
    for (; i < n; ++i) hout[i] = hin[i] * ws;
  }
}

__global__ void __launch_bounds__(THREADS)
sage_scatter(const float* __restrict__ hin, float* __restrict__ hout,
             const int* __restrict__ src, const int* __restrict__ dst,
             const float* __restrict__ w_neigh, int hop, long n_edges) {
  __shared__ StageBuf sb;
  const float wn = w_neigh[hop];                       // uniform -> s_load
  const unsigned lane = threadIdx.x & 31u;
  const unsigned w    = threadIdx.x >> 5;
  const long gw = (long)blockIdx.x * WAVES_PER_BLOCK + w;   // global wave id
  const long nw = (long)gridDim.x * WAVES_PER_BLOCK;        // total waves
  const long n_stages = n_edges >> 7;                       // full 128-edge stages
  const size_t step = (size_t)nw * STAGE_BYTES;             // bytes per wave-stride

  // LDS byte offsets (generic LDS-aperture pointer: low 32 bits = LDS offset)
  unsigned ls_cur = (unsigned)(size_t)&sb.d[0][w][0][lane];
  unsigned ld_cur = (unsigned)(size_t)&sb.d[0][w][1][lane];
  unsigned ls_nxt = (unsigned)(size_t)&sb.d[1][w][0][lane];
  unsigned ld_nxt = (unsigned)(size_t)&sb.d[1][w][1][lane];

  // Per-lane streaming pointers; advance by `step` each stage (no per-iter mul).
  const char* ps = (const char*)src + (size_t)gw * STAGE_BYTES + (size_t)lane * 16;
  const char* pd = (const char*)dst + (size_t)gw * STAGE_BYTES + (size_t)lane * 16;

  long s = gw;
  unsigned buf = 0;
  if (s < n_stages) { async_ld_b128(ls_cur, ps); async_ld_b128(ld_cur, pd); }

  while (s < n_stages) {
    const long snext = s + nw;
    ps += step; pd += step;
    if (snext < n_stages) {
      async_ld_b128(ls_nxt, ps);
      async_ld_b128(ld_nxt, pd);
      // 4 async loads outstanding; wait until only the 2 newest remain.
      asm volatile("s_wait_asynccnt 2" ::: "memory");
    } else {
      asm volatile("s_wait_asynccnt 0" ::: "memory");
    }

    // Consume current stage from LDS: 4 edges per lane (2x ds_load_b128).
    const int4 s4 = sb.d[buf][w][0][lane];
    const int4 d4 = sb.d[buf][w][1][lane];
    const float v0 = hin[s4.x] * wn;
    const float v1 = hin[s4.y] * wn;
    const float v2 = hin[s4.z] * wn;
    const float v3 = hin[s4.w] * wn;
    atomic_add_f32_dev(&hout[d4.x], v0);
    atomic_add_f32_dev(&hout[d4.y], v1);
    atomic_add_f32_dev(&hout[d4.z], v2);
    atomic_add_f32_dev(&hout[d4.w], v3);

    s = snext; buf ^= 1;
    unsigned t;
    t = ls_cur; ls_cur = ls_nxt; ls_nxt = t;
    t = ld_cur; ld_cur = ld_nxt; ld_nxt = t;
  }

  // Tail edges (n_edges % 128) — plain path, normally empty (32M % 128 == 0).
  const long base = n_stages * EDGES_PER_STAGE;
  for (long e = base + (long)blockIdx.x * THREADS + threadIdx.x; e < n_edges;
       e += (long)gridDim.x * THREADS) {
    atomic_add_f32_dev(&hout[dst[e]], hin[src[e]] * wn);
  }
}

extern "C" void kernel_launch(void* const* d_in, const int* in_sizes, int n_in,
                              void* d_out, int out_size, void* d_ws, size_t ws_size,
                              hipStream_t stream) {
  const float* h       = (const float*)d_in[0];
  const int*   src     = (const int*)  d_in[1];
  const int*   dst     = (const int*)  d_in[2];
  const float* w_self  = (const float*)d_in[3];
  const float* w_neigh = (const float*)d_in[4];
  const int  n_nodes = in_sizes[0];
  const long n_edges = (long)in_sizes[1];
  const int  num_hop = in_sizes[3];

  float* out  = (float*)d_out;
  float* bufA = (float*)d_ws;            // 2 ping-pong node buffers in scratch
  float* bufB = bufA + n_nodes;

  const int init_blocks = (n_nodes / 4 + THREADS - 1) / THREADS;
  const int scat_blocks = 4096;          // 32768 waves, ~8 stages each -> deep pipeline

  const float* hin = h;
  for (int i = 0; i < num_hop; ++i) {
    float* hout = (i == num_hop - 1) ? out : ((i & 1) ? bufB : bufA);
    sage_init_scale<<<init_blocks, THREADS, 0, stream>>>(hin, hout, w_self, i, n_nodes);
    sage_scatter<<<scat_blocks, THREADS, 0, stream>>>(hin, hout, src, dst,
                                                      w_neigh, i, n_edges);
    hin = hout;
  }
}